// GRU_24077586661462
// MI455X (gfx1250) — compile-verified
//
#include <hip/hip_runtime.h>
#include <math.h>

typedef __attribute__((ext_vector_type(2))) float v2f;
typedef __attribute__((ext_vector_type(8))) float v8f;

#define B_   128
#define T_   512
#define E_   32
#define H_   128
#define VO_  2000
#define HSTR 132   // padded LDS row stride (floats): bank-conflict-free b64 loads

__device__ __forceinline__ v2f mk2(float a, float b) { v2f r; r.x = a; r.y = b; return r; }

__device__ __forceinline__ v8f wmma4(v2f a, v2f b, v8f c) {
  // D = A(16x4,f32) * B(4x16,f32) + C(16x16,f32)
  return __builtin_amdgcn_wmma_f32_16x16x4_f32(false, a, false, b, (short)0, c, false, false);
}

__device__ __forceinline__ float sigmoidf_(float x) {
  return 1.0f / (1.0f + __expf(-x));
}

// ---------------------------------------------------------------------------
// Kernel 1: sequential GRU scan. 8 blocks (one per 16 batch rows), 8 waves
// each (one per 16-column tile of H). U/W B-fragments held in registers,
// h state in LDS. Writes h_t for all t to h_all [B,T,H] and h_last [B,H].
// ---------------------------------------------------------------------------
__global__ __launch_bounds__(256, 1)
void gru_scan_kernel(const float* __restrict__ x,
                     const float* __restrict__ W_r, const float* __restrict__ b_r,
                     const float* __restrict__ U_r, const float* __restrict__ c_r,
                     const float* __restrict__ W_z, const float* __restrict__ b_z,
                     const float* __restrict__ U_z, const float* __restrict__ c_z,
                     const float* __restrict__ W_h, const float* __restrict__ b_h,
                     const float* __restrict__ U_h, const float* __restrict__ c_h,
                     float* __restrict__ h_all,   // [B, T, H]
                     float* __restrict__ h_last)  // [B, H]
{
  __shared__ float hbuf[16 * HSTR];

  const int tid  = threadIdx.x;
  const int wave = tid >> 5;
  const int lane = tid & 31;
  const int half = lane >> 4;      // 0: K-pair {0,1} / rows 0-7 of C; 1: {2,3} / rows 8-15
  const int l15  = lane & 15;
  const int koff = half * 2;
  const int c    = wave * 16 + l15;     // H column owned by this lane
  const int b0   = blockIdx.x * 16;     // first batch row of this block

  // Per-column constants (same for all 8 C-elements of a lane).
  const float bcR = b_r[c] + c_r[c];
  const float bcZ = b_z[c] + c_z[c];
  const float bH1 = b_h[c];
  const float cH2 = c_h[c];

  // W_* B-fragments: K=E=32 -> 8 WMMA steps per gate.
  v2f bwR[8], bwZ[8], bwH[8];
#pragma unroll
  for (int j = 0; j < 8; ++j) {
    const int k0 = 4 * j + koff;
    bwR[j] = mk2(W_r[k0 * H_ + c], W_r[(k0 + 1) * H_ + c]);
    bwZ[j] = mk2(W_z[k0 * H_ + c], W_z[(k0 + 1) * H_ + c]);
    bwH[j] = mk2(W_h[k0 * H_ + c], W_h[(k0 + 1) * H_ + c]);
  }
  // U_* B-fragments: K=H=128 -> 32 WMMA steps per gate (held in VGPRs all scan).
  v2f buR[32], buZ[32], buH[32];
#pragma unroll
  for (int k = 0; k < 32; ++k) {
    const int k0 = 4 * k + koff;
    buR[k] = mk2(U_r[k0 * H_ + c], U_r[(k0 + 1) * H_ + c]);
    buZ[k] = mk2(U_z[k0 * H_ + c], U_z[(k0 + 1) * H_ + c]);
    buH[k] = mk2(U_h[k0 * H_ + c], U_h[(k0 + 1) * H_ + c]);
  }

  // h0 = 0
  for (int i = tid; i < 16 * HSTR; i += 256) hbuf[i] = 0.0f;
  __syncthreads();

  const float* xrow = x + (size_t)(b0 + l15) * T_ * E_ + koff;

  for (int t = 0; t < T_; ++t) {
    // A-fragments from x_t (shared by all three gates).
    v2f ax[8];
    const float* xp = xrow + (size_t)t * E_;
#pragma unroll
    for (int j = 0; j < 8; ++j) ax[j] = *(const v2f*)(xp + 4 * j);

    v8f accR, accZ, accH1, accH2;
#pragma unroll
    for (int v = 0; v < 8; ++v) {
      accR[v] = bcR; accZ[v] = bcZ; accH1[v] = bH1; accH2[v] = cH2;
    }

    // x @ W_* (+biases already in C)
#pragma unroll
    for (int j = 0; j < 8; ++j) {
      accR  = wmma4(ax[j], bwR[j], accR);
      accZ  = wmma4(ax[j], bwZ[j], accZ);
      accH1 = wmma4(ax[j], bwH[j], accH1);
    }
    // h @ U_* ; A-fragment of h reused across the three gates.
#pragma unroll
    for (int k = 0; k < 32; ++k) {
      v2f ah = *(const v2f*)(&hbuf[l15 * HSTR + 4 * k + koff]);
      accR  = wmma4(ah, buR[k], accR);
      accZ  = wmma4(ah, buZ[k], accZ);
      accH2 = wmma4(ah, buH[k], accH2);
    }

    float hn[8];
#pragma unroll
    for (int v = 0; v < 8; ++v) {
      const float hold = hbuf[(v + half * 8) * HSTR + c];
      const float r  = sigmoidf_(accR[v]);
      const float z  = sigmoidf_(accZ[v]);
      const float hp = tanhf(accH1[v] + r * accH2[v]);
      hn[v] = hold + z * (hp - hold);   // (1-z)*h + z*h'
    }

    __syncthreads();   // all reads of hbuf done
#pragma unroll
    for (int v = 0; v < 8; ++v) {
      const int row = v + half * 8;
      hbuf[row * HSTR + c] = hn[v];
      h_all[((size_t)(b0 + row) * T_ + t) * H_ + c] = hn[v];
    }
    if (t == T_ - 1) {
#pragma unroll
      for (int v = 0; v < 8; ++v) {
        const int row = v + half * 8;
        h_last[(size_t)(b0 + row) * H_ + c] = hn[v];
      }
    }
    __syncthreads();   // hbuf updated for next step
  }
}

// ---------------------------------------------------------------------------
// Kernel 2: out[b,t,:] = h_all[b,t,:] @ V + d_V   as [65536,128] x [128,2000].
// Block = 256 threads (8 waves) computes a 128x80 output tile; V tile staged
// transposed in LDS; 160 f32 WMMAs per wave; NT streaming stores (524 MB).
// ---------------------------------------------------------------------------
__global__ __launch_bounds__(256, 1)
void gru_out_gemm_kernel(const float* __restrict__ h_all, // [B*T, H]
                         const float* __restrict__ Vw,    // [H, VO]
                         const float* __restrict__ dV,    // [VO]
                         float* __restrict__ out)         // [B*T, VO]
{
  __shared__ float bt[80 * HSTR];   // V tile, transposed: bt[col][k]

  const int tid  = threadIdx.x;
  const int wave = tid >> 5;
  const int lane = tid & 31;
  const int half = lane >> 4;
  const int l15  = lane & 15;
  const int koff = half * 2;

  const int m0 = blockIdx.x * 128;  // row block (b*T+t space)
  const int n0 = blockIdx.y * 80;   // col block (V_OUT space)

  // Stage V[:, n0:n0+80] transposed into LDS (coalesced global reads).
  for (int i = tid; i < 128 * 80; i += 256) {
    const int k  = i / 80;
    const int cc = i - k * 80;
    bt[cc * HSTR + k] = Vw[(size_t)k * VO_ + n0 + cc];
  }
  __syncthreads();

  v8f acc[5];
#pragma unroll
  for (int s = 0; s < 5; ++s) {
    const float dv = dV[n0 + s * 16 + l15];
#pragma unroll
    for (int v = 0; v < 8; ++v) acc[s][v] = dv;
  }

  const float* ap = h_all + (size_t)(m0 + wave * 16 + l15) * H_ + koff;
#pragma unroll
  for (int k = 0; k < 32; ++k) {
    const v2f a = *(const v2f*)(ap + 4 * k);
#pragma unroll
    for (int s = 0; s < 5; ++s) {
      const v2f b = *(const v2f*)(&bt[(s * 16 + l15) * HSTR + 4 * k + koff]);
      acc[s] = wmma4(a, b, acc[s]);
    }
  }

  const size_t rowBase = (size_t)(m0 + wave * 16 + half * 8);
#pragma unroll
  for (int v = 0; v < 8; ++v) {
    float* op = out + (rowBase + v) * VO_ + n0 + l15;
#pragma unroll
    for (int s = 0; s < 5; ++s) {
      __builtin_nontemporal_store(acc[s][v], op + s * 16);
    }
  }
}

// ---------------------------------------------------------------------------
extern "C" void kernel_launch(void* const* d_in, const int* in_sizes, int n_in,
                              void* d_out, int out_size, void* d_ws, size_t ws_size,
                              hipStream_t stream) {
  (void)in_sizes; (void)n_in; (void)out_size; (void)ws_size;

  const float* x   = (const float*)d_in[0];
  const float* W_r = (const float*)d_in[1];
  const float* b_r = (const float*)d_in[2];
  const float* U_r = (const float*)d_in[3];
  const float* c_r = (const float*)d_in[4];
  const float* W_z = (const float*)d_in[5];
  const float* b_z = (const float*)d_in[6];
  const float* U_z = (const float*)d_in[7];
  const float* c_z = (const float*)d_in[8];
  const float* W_h = (const float*)d_in[9];
  const float* b_h = (const float*)d_in[10];
  const float* U_h = (const float*)d_in[11];
  const float* c_h = (const float*)d_in[12];
  const float* Vw  = (const float*)d_in[13];
  const float* dV  = (const float*)d_in[14];

  float* out    = (float*)d_out;
  float* h_last = out + (size_t)B_ * T_ * VO_;   // outputs concatenated flat
  float* h_all  = (float*)d_ws;                  // needs B*T*H*4 = 33.5 MB

  gru_scan_kernel<<<dim3(B_ / 16), dim3(256), 0, stream>>>(
      x, W_r, b_r, U_r, c_r, W_z, b_z, U_z, c_z, W_h, b_h, U_h, c_h,
      h_all, h_last);

  gru_out_gemm_kernel<<<dim3((B_ * T_) / 128, VO_ / 80), dim3(256), 0, stream>>>(
      h_all, Vw, dV, out);
}